// YoloLayer_88553635709489
// MI455X (gfx1250) — compile-verified
//
#include <hip/hip_runtime.h>
#include <hip/hip_bf16.h>
#include <stdint.h>

// Problem constants (match reference)
#define BATCH 64
#define NA    3
#define NC    80
#define GH    76
#define GW    76
#define HW    (GH * GW)        // 5776
#define CHN   (5 + NC)         // 85 channels per anchor
#define TILE  32               // grid positions per block
#define PITCH 33               // LDS row pitch in floats (32 data + 1 pad dword from TDM pad)
#define NTILES 181             // ceil(5776/32); last tile start clamped to 5744
#define NBOX  (NA * HW)        // 17328

typedef unsigned int v4u __attribute__((ext_vector_type(4)));
typedef int          v8i __attribute__((ext_vector_type(8)));
typedef int          v4i __attribute__((ext_vector_type(4)));

__device__ __forceinline__ float fast_sigmoid(float x) {
    // 1 / (1 + e^-x) using v_exp_f32 + v_rcp_f32
    return __builtin_amdgcn_rcpf(1.0f + __expf(-x));
}

__global__ __launch_bounds__(256) void yolo_layer_kernel(const float* __restrict__ in,
                                                         float* __restrict__ out)
{
    // 85 channel rows x (32 + 1 pad) floats = 11220 bytes of LDS
    __shared__ float lds[CHN * PITCH];

    const int tile = blockIdx.x;
    const int a    = blockIdx.y;
    const int b    = blockIdx.z;

    int start = tile * TILE;
    if (start > HW - TILE) start = HW - TILE;   // clamp; overlap recomputes identical values

    const float* src = in + (size_t)(b * NA + a) * (size_t)(CHN * HW);

#if __has_builtin(__builtin_amdgcn_tensor_load_to_lds)
    if (threadIdx.x < 32) {   // wave 0 only issues the TDM op (EXEC is ignored, branch is wave-uniform)
        const uint64_t gaddr   = (uint64_t)(uintptr_t)(src + start); // tile start byte address
        const uint32_t ldsaddr = (uint32_t)(uintptr_t)(&lds[0]);     // low 32 bits = LDS byte offset

        // ---- D# group 0 (128b): count | lds_addr | global_addr | type=2 ----
        v4u g0;
        g0.x = 1u;                                                   // count=1, user descriptor
        g0.y = ldsaddr;                                              // lds_addr[31:0]
        g0.z = (uint32_t)(gaddr & 0xFFFFFFFFu);                      // global_addr[31:0]
        g0.w = (uint32_t)((gaddr >> 32) & 0x01FFFFFFu) | (2u << 30); // global_addr[56:32] | type=2

        // ---- D# group 1 (256b): 2D tensor [CHN rows x HW cols] f32, tile [CHN x TILE] ----
        // word0: workgroup_mask=0 | data_size=2 (4B) | pad_enable | pad_interval=4 (32 dwords) | pad_amount=0 (1 dword)
        v8i g1;
        g1[0] = (int)((2u << 16) | (1u << 20) | (4u << 22));
        g1[1] = (int)((uint32_t)(HW & 0xFFFF) << 16);                           // tensor_dim0[15:0]
        g1[2] = (int)((uint32_t)(HW >> 16) | ((uint32_t)(CHN & 0xFFFF) << 16)); // dim0 hi | tensor_dim1 lo
        g1[3] = (int)(((uint32_t)TILE) << 16);                                  // dim1 hi=0 | tile_dim0=32
        g1[4] = (int)((uint32_t)CHN);                                           // tile_dim1=85, tile_dim2=0
        g1[5] = (int)((uint32_t)HW);                                            // tensor_dim0_stride[31:0]
        g1[6] = 0;                                                              // stride hi | dim1_stride lo (2D: unused)
        g1[7] = 0;

        v4i g2 = {0, 0, 0, 0};                  // 2D tile: groups 2/3 unused (tile_dim2/3 = 0)
        v4i g3 = {0, 0, 0, 0};
        v8i g4 = {0, 0, 0, 0, 0, 0, 0, 0};      // trailing group (clang-23 6-arg form), unused

        __builtin_amdgcn_tensor_load_to_lds(g0, g1, g2, g3, g4, 0);
        __builtin_amdgcn_s_wait_tensorcnt(0);
    }
    __syncthreads();
#else
    // Fallback: staged coalesced loads (32 contiguous floats per channel row)
    for (int i = threadIdx.x; i < CHN * TILE; i += 256) {
        const int ch = i >> 5;
        const int p  = i & 31;
        lds[ch * PITCH + p] = src[(size_t)ch * HW + (size_t)(start + p)];
    }
    __syncthreads();
#endif

    const int wave = threadIdx.x >> 5;
    const int lane = threadIdx.x & 31;

    float* boxes = out;                                   // [B][NBOX][1][4]
    float* confs = out + (size_t)BATCH * NBOX * 4;        // [B][NBOX][NC]

    // anchors[a] = (w, h), already divided by stride in reference
    const float AW[NA] = {1.5f, 2.375f, 5.0f};
    const float AH[NA] = {2.0f, 4.5f, 3.5f};

    // ---- boxes: wave 0, one lane per grid position ----
    if (wave == 0) {
        const int   col = start + lane;
        const int   gy  = col / GW;
        const int   gx  = col - gy * GW;
        const float c0  = lds[0 * PITCH + lane];
        const float c1  = lds[1 * PITCH + lane];
        const float c2  = lds[2 * PITCH + lane];
        const float c3  = lds[3 * PITCH + lane];

        // SCALE_XY == 1.0 -> bxy = sigmoid(o)
        const float bx = (fast_sigmoid(c0) + (float)gx) * (1.0f / (float)GW);
        const float by = (fast_sigmoid(c1) + (float)gy) * (1.0f / (float)GH);
        const float bw = __expf(c2) * (AW[a] * (1.0f / (float)GW));
        const float bh = __expf(c3) * (AH[a] * (1.0f / (float)GH));

        const float x1 = bx - 0.5f * bw;
        const float y1 = by - 0.5f * bh;
        const size_t bo = ((size_t)b * NBOX + (size_t)a * HW + (size_t)col) * 4;
        *(float4*)(boxes + bo) = make_float4(x1, y1, x1 + bw, y1 + bh);
    }

    // ---- confs: each wave handles 4 positions; lanes sweep channels (coalesced 128B stores) ----
    #pragma unroll
    for (int i = 0; i < 4; ++i) {
        const int p   = wave * 4 + i;
        const int col = start + p;
        const float det = fast_sigmoid(lds[4 * PITCH + p]);   // same-address LDS broadcast
        const size_t co = ((size_t)b * NBOX + (size_t)a * HW + (size_t)col) * (size_t)NC;
        #pragma unroll
        for (int j = 0; j < 3; ++j) {
            const int cc = lane + j * 32;
            if (cc < NC) {
                confs[co + cc] = fast_sigmoid(lds[(5 + cc) * PITCH + p]) * det;
            }
        }
    }
}

extern "C" void kernel_launch(void* const* d_in, const int* in_sizes, int n_in,
                              void* d_out, int out_size, void* d_ws, size_t ws_size,
                              hipStream_t stream) {
    (void)in_sizes; (void)n_in; (void)d_ws; (void)ws_size; (void)out_size;
    const float* in = (const float*)d_in[0];
    float* out = (float*)d_out;
    dim3 grid(NTILES, NA, BATCH);
    yolo_layer_kernel<<<grid, 256, 0, stream>>>(in, out);
}